// SpectrogramLayer_7722351198624
// MI455X (gfx1250) — compile-verified
//
#include <hip/hip_runtime.h>
#include <hip/hip_bf16.h>
#include <math.h>

typedef float v2f __attribute__((ext_vector_type(2)));
typedef float v8f __attribute__((ext_vector_type(8)));

#define N_PTS    2048   // signal length n
#define N_FREQ   2049   // n+1 one-sided bins == number of frame centers (HOP=1)
#define TILE     64     // output tile edge (freq x center)
#define TK       16     // t-chunk staged per iteration (4 wmma K-steps of 4)
#define NTHREADS 128    // 4 wave32 per block

// pi / 2048
#define PI_OVER_N 1.5339807878856412e-3f

__global__ __launch_bounds__(NTHREADS)
void spectro_wmma_kernel(const float* __restrict__ x,
                         const float* __restrict__ lambd,
                         float* __restrict__ out)
{
    __shared__ float xLds[N_PTS];          // centered signal row for this batch
    __shared__ float sred[NTHREADS];
    __shared__ float AsC[TILE][TK + 1];    // cos(pi*k*t/n), padded rows (17 banks-coprime)
    __shared__ float AsS[TILE][TK + 1];    // sin(pi*k*t/n)
    __shared__ float Bs[TK][TILE + 1];     // x[t]*w(c,t)

    const int tid    = threadIdx.x;
    const int b      = blockIdx.z;
    const int kfBase = blockIdx.y * TILE;  // frequency-bin base of this tile
    const int cBase  = blockIdx.x * TILE;  // frame-center base of this tile

    const float sigma     = fabsf(lambd[0]);
    const float inv_sigma = 1.0f / sigma;

    // ---- load x[b,:] into LDS, compute mean, subtract (per-batch centering) ----
    float partial = 0.0f;
    for (int i = tid; i < N_PTS; i += NTHREADS) {
        float v = x[(size_t)b * N_PTS + i];
        xLds[i] = v;
        partial += v;
    }
    sred[tid] = partial;
    __syncthreads();
    for (int s = NTHREADS / 2; s > 0; s >>= 1) {
        if (tid < s) sred[tid] += sred[tid + s];
        __syncthreads();
    }
    const float mean = sred[0] * (1.0f / (float)N_PTS);
    for (int i = tid; i < N_PTS; i += NTHREADS) xLds[i] -= mean;

    // ---- per-lane fragment coordinates (ISA 7.12.2, f32 16x16x4 layouts) ----
    const int lane = tid & 31;
    const int wv   = tid >> 5;          // wave id 0..3: owns freq rows [wv*16, wv*16+16)
    const int half = lane >> 4;         // lane block 0/1
    const int mrow = lane & 15;         // A row (M) / B,C column (N) within tile
    const int koff = half << 1;         // A: lanes 0-15 hold K={kk,kk+1}, 16-31 hold K={kk+2,kk+3}
    const int arow = (wv << 4) + mrow;

    v8f accRe[4], accIm[4];
    #pragma unroll
    for (int nt = 0; nt < 4; ++nt) {
        #pragma unroll
        for (int e = 0; e < 8; ++e) { accRe[nt][e] = 0.0f; accIm[nt][e] = 0.0f; }
    }

    // ---- main K-loop over signal time t ----
    for (int t0 = 0; t0 < N_PTS; t0 += TK) {
        __syncthreads();   // previous-iteration LDS reads done before refill

        // stage A = DFT twiddles for this tile's 64 freq rows x 16 t-columns.
        // reduce k*t mod 4096 in exact integer math -> angle in [-pi, pi)
        #pragma unroll
        for (int i = 0; i < (TILE * TK) / NTHREADS; ++i) {   // 8 elems/thread
            int e   = tid + i * NTHREADS;
            int kfl = e >> 4;
            int tt  = e & (TK - 1);
            int kg  = kfBase + kfl;
            int tg  = t0 + tt;
            int mm  = (kg * tg) & 4095;
            if (mm >= 2048) mm -= 4096;
            float ang = (float)mm * PI_OVER_N;
            float sv, cv;
            __sincosf(ang, &sv, &cv);
            AsC[kfl][tt] = cv;
            AsS[kfl][tt] = sv;
        }
        // stage B = centered-signal * Gaussian window, 16 t-rows x 64 center-cols
        #pragma unroll
        for (int i = 0; i < (TK * TILE) / NTHREADS; ++i) {   // 8 elems/thread
            int e  = tid + i * NTHREADS;
            int tt = e >> 6;
            int cl = e & 63;
            int tg = t0 + tt;
            float d = (float)(tg - (cBase + cl)) * inv_sigma;
            Bs[tt][cl] = xLds[tg] * __expf(-0.5f * d * d);
        }
        __syncthreads();

        // 4 K-steps of 4, 4 center-tiles, Re+Im accumulators sharing B
        #pragma unroll
        for (int kk = 0; kk < TK; kk += 4) {
            v2f ac, as;
            ac.x = AsC[arow][kk + koff];
            ac.y = AsC[arow][kk + koff + 1];
            as.x = AsS[arow][kk + koff];
            as.y = AsS[arow][kk + koff + 1];
            #pragma unroll
            for (int nt = 0; nt < 4; ++nt) {
                v2f bb;
                bb.x = Bs[kk + koff][nt * 16 + mrow];
                bb.y = Bs[kk + koff + 1][nt * 16 + mrow];
                accRe[nt] = __builtin_amdgcn_wmma_f32_16x16x4_f32(
                    false, ac, false, bb, (short)0, accRe[nt], false, false);
                accIm[nt] = __builtin_amdgcn_wmma_f32_16x16x4_f32(
                    false, as, false, bb, (short)0, accIm[nt], false, false);
            }
        }
    }

    // ---- power spectrum + store: out[b, 0, kf, c], C/D layout VGPR r -> M=r (+8 for lanes>=16) ----
    const int kfW = kfBase + wv * 16;
    #pragma unroll
    for (int nt = 0; nt < 4; ++nt) {
        int cg = cBase + nt * 16 + mrow;
        if (cg < N_FREQ) {
            #pragma unroll
            for (int r = 0; r < 8; ++r) {
                int kf = kfW + r + 8 * half;
                if (kf < N_FREQ) {
                    float re = accRe[nt][r];
                    float im = accIm[nt][r];
                    out[((size_t)b * N_FREQ + kf) * (size_t)N_FREQ + cg] = re * re + im * im;
                }
            }
        }
    }
}

extern "C" void kernel_launch(void* const* d_in, const int* in_sizes, int n_in,
                              void* d_out, int out_size, void* d_ws, size_t ws_size,
                              hipStream_t stream) {
    (void)in_sizes; (void)n_in; (void)d_ws; (void)ws_size; (void)out_size;
    const float* x     = (const float*)d_in[0];
    const float* lambd = (const float*)d_in[1];
    float* out = (float*)d_out;

    dim3 grid((N_FREQ + TILE - 1) / TILE,   // 33 center tiles
              (N_FREQ + TILE - 1) / TILE,   // 33 freq tiles
              8);                           // batch
    dim3 block(NTHREADS, 1, 1);
    spectro_wmma_kernel<<<grid, block, 0, stream>>>(x, lambd, out);
}